// AdaptiveSSM_76132590289374
// MI455X (gfx1250) — compile-verified
//
#include <hip/hip_runtime.h>
#include <hip/hip_bf16.h>
#include <math.h>

typedef __attribute__((ext_vector_type(2))) float v2f;
typedef __attribute__((ext_vector_type(4))) float v4f;
typedef __attribute__((ext_vector_type(8))) float v8f;

#define HID 128
#define STATE 16

__global__ __launch_bounds__(256, 1) void AdaptiveSSM_fused_wmma(
    const float* __restrict__ h_t,      // [N, HID]
    const float* __restrict__ u_prev,   // [N, HID, STATE]
    const float* __restrict__ W_delta,  // [HID, HID] row-major
    const float* __restrict__ b_delta,  // [HID]
    const float* __restrict__ log_A,    // [HID, STATE]
    const float* __restrict__ Bmat,     // [HID, STATE]
    const float* __restrict__ Cmat,     // [HID, STATE]
    float* __restrict__ y_out,          // [N, HID]
    float* __restrict__ u_out,          // [N, HID, STATE]
    int N)
{
    __shared__ float sNegA[HID * STATE];   // -exp(log_A)
    __shared__ float sB[HID * STATE];
    __shared__ float sC[HID * STATE];

    const int tid = threadIdx.x;
    for (int i = tid; i < HID * STATE; i += 256) {
        sNegA[i] = -__expf(log_A[i]);
        sB[i]    = Bmat[i];
        sC[i]    = Cmat[i];
    }
    __syncthreads();

    const int row0 = blockIdx.x * 16;
    const int wave = tid >> 5;     // 8 waves, each owns a 16-wide j-tile
    const int lane = tid & 31;
    const int half = lane >> 4;    // 0: lanes 0-15, 1: lanes 16-31
    const int l16  = lane & 15;
    const int j0   = wave * 16;

    // Clamp A-row so EXEC stays all-ones through the WMMA sequence.
    int arow = row0 + l16;
    if (arow >= N) arow = N - 1;

    // A frag: lane m loads h[m][k + half*2 .. +1]  (ISA 16x4 f32 A layout)
    // B frag: lane n loads W_delta[j0+n][k + half*2 .. +1] (B[k][n] = W^T)
    const float* aBase = h_t     + (size_t)arow * HID        + (half << 1);
    const float* bBase = W_delta + (size_t)(j0 + l16) * HID  + (half << 1);

    v8f acc = {};
#pragma unroll
    for (int k = 0; k < HID; k += 4) {
        v2f a = *(const v2f*)(aBase + k);
        v2f b = *(const v2f*)(bBase + k);
        acc = __builtin_amdgcn_wmma_f32_16x16x4_f32(
            /*neg_a=*/false, a, /*neg_b=*/false, b,
            /*c_mod=*/(short)0, acc, /*reuse_a=*/false, /*reuse_b=*/false);
    }

    // C/D layout: VGPR v -> M = v + half*8, N = l16
    const int   j    = j0 + l16;
    const float bias = b_delta[j];

    // Per-lane parameter registers (same j for all 8 rows this lane owns)
    float pA[STATE], pB[STATE], pC[STATE];
#pragma unroll
    for (int s = 0; s < STATE; ++s) {
        pA[s] = sNegA[j * STATE + s];
        pB[s] = sB[j * STATE + s];
        pC[s] = sC[j * STATE + s];
    }

#pragma unroll
    for (int v = 0; v < 8; ++v) {
        const int m   = v + (half << 3);
        const int row = row0 + m;
        if (row >= N) continue;   // guard stores only; WMMA already done

        const float logit = acc[v] + bias;
        // stable softplus: max(x,0) + log1p(exp(-|x|))
        const float delta = fmaxf(logit, 0.0f) + log1pf(__expf(-fabsf(logit)));
        const float hv    = h_t[(size_t)row * HID + j];
        const float dh    = delta * hv;

        const size_t base = ((size_t)row * HID + j) * STATE;
        const v4f* up = (const v4f*)(u_prev + base);
        v4f*       uo = (v4f*)(u_out + base);

        float y = 0.0f;
#pragma unroll
        for (int q = 0; q < 4; ++q) {
            const int s = q * 4;
            // u stream is 820 MB (>> 192 MB L2): non-temporal so L2 keeps
            // W_delta / h_t / params instead.
            v4f u4 = __builtin_nontemporal_load(up + q);
            v4f r;
            r.x = u4.x * __expf(delta * pA[s + 0]) + dh * pB[s + 0];
            r.y = u4.y * __expf(delta * pA[s + 1]) + dh * pB[s + 1];
            r.z = u4.z * __expf(delta * pA[s + 2]) + dh * pB[s + 2];
            r.w = u4.w * __expf(delta * pA[s + 3]) + dh * pB[s + 3];
            y += r.x * pC[s + 0] + r.y * pC[s + 1]
               + r.z * pC[s + 2] + r.w * pC[s + 3];
            __builtin_nontemporal_store(r, uo + q);
        }
        __builtin_nontemporal_store(y, y_out + (size_t)row * HID + j);
    }
}

extern "C" void kernel_launch(void* const* d_in, const int* in_sizes, int n_in,
                              void* d_out, int out_size, void* d_ws, size_t ws_size,
                              hipStream_t stream) {
    const float* h_t     = (const float*)d_in[0];
    const float* u_prev  = (const float*)d_in[1];
    const float* W_delta = (const float*)d_in[2];
    const float* b_delta = (const float*)d_in[3];
    const float* log_A   = (const float*)d_in[4];
    const float* Bmat    = (const float*)d_in[5];
    const float* Cmat    = (const float*)d_in[6];

    const int N = in_sizes[0] / HID;            // 50000
    float* y_out = (float*)d_out;               // y_t: N*HID
    float* u_out = y_out + (size_t)N * HID;     // u_t: N*HID*STATE (concatenated)

    const int grid = (N + 15) / 16;             // 16 rows per block
    AdaptiveSSM_fused_wmma<<<grid, 256, 0, stream>>>(
        h_t, u_prev, W_delta, b_delta, log_A, Bmat, Cmat, y_out, u_out, N);
}